// freebase_des_hnode_prompt_layer_feature_weighted_sum_21534966022311
// MI455X (gfx1250) — compile-verified
//
#include <hip/hip_runtime.h>
#include <hip/hip_bf16.h>
#include <cstdint>

// Problem constants (reference: N=50000, D=128, E=640000)
#define DD   128
#define D4   32          // DD / 4 float4s per row
#define EPB  256         // edges per block (one TDM tile)
#define WPB  8           // waves per 256-thread block (wave32)

typedef unsigned int v4u __attribute__((ext_vector_type(4)));
typedef int          v8i __attribute__((ext_vector_type(8)));
typedef int          v4i __attribute__((ext_vector_type(4)));

__device__ __forceinline__ float elu1(float x) {
    return x > 0.0f ? x : (expf(x) - 1.0f);
}

// ---------------------------------------------------------------------------
// Kernel A: emb = elu(ge * w) -> workspace; zero the output in the same pass.
// Pure streaming pass, float4 (global_load_b128 / global_store_b128).
// ---------------------------------------------------------------------------
__global__ void prep_elu_zero(const float4* __restrict__ ge,
                              const float4* __restrict__ w4,
                              float4* __restrict__ emb,
                              float4* __restrict__ out,
                              int n4) {
    int i = blockIdx.x * blockDim.x + threadIdx.x;
    if (i >= n4) return;
    float4 x = ge[i];
    float4 w = w4[i & (D4 - 1)];
    float4 r;
    r.x = elu1(x.x * w.x);
    r.y = elu1(x.y * w.y);
    r.z = elu1(x.z * w.z);
    r.w = elu1(x.w * w.w);
    emb[i] = r;
    out[i] = make_float4(0.f, 0.f, 0.f, 0.f);
}

__global__ void zero_out(float4* __restrict__ out, int n4) {
    int i = blockIdx.x * blockDim.x + threadIdx.x;
    if (i < n4) out[i] = make_float4(0.f, 0.f, 0.f, 0.f);
}

// ---------------------------------------------------------------------------
// TDM: 1-D tile of int32 (count "valid" elements in-bounds, tile of EPB) from
// global memory into LDS at byte offset lds_off. Descriptor per ISA ch.8:
//   group0: count=1 | lds_addr | global_addr[56:0] | type=2
//   group1: data_size=2 (4B) | tensor_dim0=valid (OOB reads -> 0)
//           | tensor_dim1=1 | tile_dim0=EPB | tensor_dim0_stride=EPB
//   groups 2/3: zero (<=2D tensor)
// ---------------------------------------------------------------------------
__device__ __forceinline__ void tdm_load_tile_i32(uint32_t lds_off,
                                                  const int* gptr,
                                                  int valid, int tile) {
    uint64_t ga = (uint64_t)(uintptr_t)gptr;
    v4u g0;
    g0[0] = 1u;                                              // count=1, user desc
    g0[1] = lds_off;                                         // lds_addr
    g0[2] = (uint32_t)ga;                                    // global_addr lo
    g0[3] = (uint32_t)((ga >> 32) & 0x1FFFFFFu) | (2u << 30);// addr hi | type=2
    v8i g1;
    g1[0] = (int)(2u << 16);                                 // data_size = 4B
    g1[1] = (int)(((uint32_t)valid & 0xFFFFu) << 16);        // tensor_dim0 lo16
    g1[2] = (int)((((uint32_t)valid) >> 16) | (1u << 16));   // dim0 hi16 | dim1=1
    g1[3] = (int)(((uint32_t)tile) << 16);                   // tile_dim0
    g1[4] = 0;                                               // tile_dim1/2 unused
    g1[5] = tile;                                            // dim0_stride lo32
    g1[6] = 0;
    g1[7] = 0;
    v4i z4 = {0, 0, 0, 0};
#if __clang_major__ >= 23
    v8i z8 = {0, 0, 0, 0, 0, 0, 0, 0};
    __builtin_amdgcn_tensor_load_to_lds(g0, g1, z4, z4, z8, 0);
#else
    __builtin_amdgcn_tensor_load_to_lds(g0, g1, z4, z4, 0);
#endif
}

// ---------------------------------------------------------------------------
// Kernel B: per block, TDM-stage EPB edges (src,dst,e) into LDS, then
// one wave per edge: lane l gathers emb[src][4l..4l+3] (one b128 load,
// L2-resident) and scatters 4 f32 atomics into out[dst].
// ---------------------------------------------------------------------------
__global__ void edge_scatter(const float4* __restrict__ emb,
                             const int* __restrict__ src,
                             const int* __restrict__ dst,
                             const int* __restrict__ ef,
                             float* __restrict__ out,
                             int n_edges) {
    __shared__ int s_idx[3 * EPB];

    int base  = blockIdx.x * EPB;
    int valid = n_edges - base;
    if (valid > EPB) valid = EPB;

    if (threadIdx.x < 32) {  // wave 0 drives the TDM (EXEC ignored by TDM)
        uint32_t lb = (uint32_t)(uintptr_t)(&s_idx[0]);
        tdm_load_tile_i32(lb,               src + base, valid, EPB);
        tdm_load_tile_i32(lb + EPB * 4,     dst + base, valid, EPB);
        tdm_load_tile_i32(lb + 2 * EPB * 4, ef  + base, valid, EPB);
        __builtin_amdgcn_s_wait_tensorcnt(0);
    }
    __syncthreads();

    int wave = threadIdx.x >> 5;
    int lane = threadIdx.x & 31;
    int kbeg = wave * (EPB / WPB);
    int kend = kbeg + (EPB / WPB);
    if (kend > valid) kend = valid;

    for (int k = kbeg; k < kend; ++k) {
        int s = s_idx[k];
        int d = s_idx[EPB + k];
        int e = s_idx[2 * EPB + k];
        if (k + 1 < kend)  // warm next gather row (global_prefetch_b8)
            __builtin_prefetch(&emb[(size_t)s_idx[k + 1] * D4], 0, 0);
        float m = ((e == 0) | (e == 6) | (e == 14) | (e == 30)) ? 2.0f : 1.0f;
        float4 v = emb[(size_t)s * D4 + lane];
        float* o = out + (size_t)d * DD + lane * 4;
        atomicAdd(o + 0, v.x * m);
        atomicAdd(o + 1, v.y * m);
        atomicAdd(o + 2, v.z * m);
        atomicAdd(o + 3, v.w * m);
    }
}

// Fallback if workspace is too small: recompute elu(x*w) at gather time.
__global__ void edge_scatter_fused(const float4* __restrict__ ge,
                                   const float4* __restrict__ w4,
                                   const int* __restrict__ src,
                                   const int* __restrict__ dst,
                                   const int* __restrict__ ef,
                                   float* __restrict__ out,
                                   int n_edges) {
    __shared__ int s_idx[3 * EPB];

    int base  = blockIdx.x * EPB;
    int valid = n_edges - base;
    if (valid > EPB) valid = EPB;

    if (threadIdx.x < 32) {
        uint32_t lb = (uint32_t)(uintptr_t)(&s_idx[0]);
        tdm_load_tile_i32(lb,               src + base, valid, EPB);
        tdm_load_tile_i32(lb + EPB * 4,     dst + base, valid, EPB);
        tdm_load_tile_i32(lb + 2 * EPB * 4, ef  + base, valid, EPB);
        __builtin_amdgcn_s_wait_tensorcnt(0);
    }
    __syncthreads();

    int wave = threadIdx.x >> 5;
    int lane = threadIdx.x & 31;
    float4 w = w4[lane];  // lane owns columns [4*lane, 4*lane+4)

    int kbeg = wave * (EPB / WPB);
    int kend = kbeg + (EPB / WPB);
    if (kend > valid) kend = valid;

    for (int k = kbeg; k < kend; ++k) {
        int s = s_idx[k];
        int d = s_idx[EPB + k];
        int e = s_idx[2 * EPB + k];
        float m = ((e == 0) | (e == 6) | (e == 14) | (e == 30)) ? 2.0f : 1.0f;
        float4 x = ge[(size_t)s * D4 + lane];
        float* o = out + (size_t)d * DD + lane * 4;
        atomicAdd(o + 0, elu1(x.x * w.x) * m);
        atomicAdd(o + 1, elu1(x.y * w.y) * m);
        atomicAdd(o + 2, elu1(x.z * w.z) * m);
        atomicAdd(o + 3, elu1(x.w * w.w) * m);
    }
}

extern "C" void kernel_launch(void* const* d_in, const int* in_sizes, int n_in,
                              void* d_out, int out_size, void* d_ws, size_t ws_size,
                              hipStream_t stream) {
    const float* ge = (const float*)d_in[0];   // [N, 128] f32
    const float* w  = (const float*)d_in[1];   // [1, 128] f32
    const int*   ef = (const int*)d_in[2];     // [E] int
    const int*   sr = (const int*)d_in[3];     // [E] int
    const int*   ds = (const int*)d_in[4];     // [E] int
    float* out = (float*)d_out;                // [N, 128] f32

    int nD      = in_sizes[0];                 // N*D
    int n4      = nD / 4;
    int n_edges = in_sizes[2];
    int pblocks = (n4 + 255) / 256;
    int eblocks = (n_edges + EPB - 1) / EPB;

    size_t need = (size_t)nD * sizeof(float);
    if (ws_size >= need) {
        float* emb = (float*)d_ws;
        prep_elu_zero<<<pblocks, 256, 0, stream>>>(
            (const float4*)ge, (const float4*)w, (float4*)emb, (float4*)out, n4);
        edge_scatter<<<eblocks, 256, 0, stream>>>(
            (const float4*)emb, sr, ds, ef, out, n_edges);
    } else {
        zero_out<<<pblocks, 256, 0, stream>>>((float4*)out, n4);
        edge_scatter_fused<<<eblocks, 256, 0, stream>>>(
            (const float4*)ge, (const float4*)w, sr, ds, ef, out, n_edges);
    }
}